// SequenceRelationshipAttention_3951369912474
// MI455X (gfx1250) — compile-verified
//
#include <hip/hip_runtime.h>

typedef __attribute__((ext_vector_type(2))) float v2f;
typedef __attribute__((ext_vector_type(8))) float v8f;

#define D_MODEL 1024
#define NB 4
#define SEQ 1024
#define NH 16
#define DK 64
#define NEG_INF (-3.402823466e38f)

static __device__ __forceinline__ v8f wmma4(v2f a, v2f b, v8f c) {
  // D = A(16x4,f32) x B(4x16,f32) + C(16x16,f32)
  return __builtin_amdgcn_wmma_f32_16x16x4_f32(false, a, false, b, (short)0, c, false, false);
}

static __device__ __forceinline__ v2f ld2(const float* p) { return *(const v2f*)p; }

// reduce across the 16 lanes of a half-wave (xor masks 1,2,4,8 stay in-half)
static __device__ __forceinline__ float half16_max(float v) {
  v = fmaxf(v, __shfl_xor(v, 1, 32));
  v = fmaxf(v, __shfl_xor(v, 2, 32));
  v = fmaxf(v, __shfl_xor(v, 4, 32));
  v = fmaxf(v, __shfl_xor(v, 8, 32));
  return v;
}
static __device__ __forceinline__ float half16_sum(float v) {
  v += __shfl_xor(v, 1, 32);
  v += __shfl_xor(v, 2, 32);
  v += __shfl_xor(v, 4, 32);
  v += __shfl_xor(v, 8, 32);
  return v;
}

// C[m,n] = sum_k A[m,k] * B[n,k]  (NT GEMM, both row-major contiguous in k)
// One wave computes a 32x32 tile as 2x2 WMMA accumulators.
static __device__ __forceinline__ void gemm32x32_nt(
    const float* __restrict__ A, const float* __restrict__ B,
    int m0, int n0, int lane, int K, v8f acc[2][2]) {
  int lm = lane & 15;
  int ko = (lane < 16) ? 0 : 2;   // A-frag: K = {ko, ko+1} per ISA 16x4 layout
  const float* a0p = A + (size_t)(m0 + lm) * K + ko;
  const float* a1p = A + (size_t)(m0 + 16 + lm) * K + ko;
  const float* b0p = B + (size_t)(n0 + lm) * K + ko;
  const float* b1p = B + (size_t)(n0 + 16 + lm) * K + ko;
#pragma unroll 4
  for (int kk = 0; kk < K; kk += 4) {
    v2f a0 = ld2(a0p + kk);
    v2f a1 = ld2(a1p + kk);
    v2f b0 = ld2(b0p + kk);
    v2f b1 = ld2(b1p + kk);
    acc[0][0] = wmma4(a0, b0, acc[0][0]);
    acc[0][1] = wmma4(a0, b1, acc[0][1]);
    acc[1][0] = wmma4(a1, b0, acc[1][0]);
    acc[1][1] = wmma4(a1, b1, acc[1][1]);
  }
}

// ---------------- Kernel 1: QKV projection ----------------
// y[m,e] = sum_d x[m,d]*W[e,d] + b[e], scattered to [b,h,s,dk] head-major.
__global__ __launch_bounds__(256) void qkv_proj_kernel(
    const float* __restrict__ x,
    const float* __restrict__ Wq, const float* __restrict__ bq,
    const float* __restrict__ Wk, const float* __restrict__ bk,
    const float* __restrict__ Wv, const float* __restrict__ bv,
    float* __restrict__ q, float* __restrict__ k, float* __restrict__ v) {
  int lane = threadIdx.x & 31;
  int tile = blockIdx.x * 8 + (threadIdx.x >> 5);
  int tM = tile >> 5;  // 128 M-tiles (M = 4096)
  int tN = tile & 31;  // 32 N-tiles (N = 1024)
  int m0 = tM * 32, n0 = tN * 32;
  const float* W; const float* bias; float* out;
  if (blockIdx.y == 0)      { W = Wq; bias = bq; out = q; }
  else if (blockIdx.y == 1) { W = Wk; bias = bk; out = k; }
  else                      { W = Wv; bias = bv; out = v; }
  v8f acc[2][2] = {};
  gemm32x32_nt(x, W, m0, n0, lane, D_MODEL, acc);
  int lm = lane & 15;
  int half8 = (lane < 16) ? 0 : 8;
#pragma unroll
  for (int i = 0; i < 2; ++i)
#pragma unroll
    for (int j = 0; j < 2; ++j) {
      int n = n0 + j * 16 + lm;
      float bval = bias[n];
      int hh = n >> 6, dd = n & 63;
#pragma unroll
      for (int r = 0; r < 8; ++r) {
        int m = m0 + i * 16 + r + half8;
        int bb = m >> 10, ss = m & 1023;
        out[((size_t)((bb * NH + hh) << 10) + ss) * DK + dd] = acc[i][j][r] + bval;
      }
    }
}

// ---------------- Kernel 2: flash attention ----------------
// One wave per (b, h, 16-query block). Online softmax, WMMA for QK^T and PV.
__global__ __launch_bounds__(128) void attn_kernel(
    const float* __restrict__ q, const float* __restrict__ kmat,
    const float* __restrict__ vmat, const int* __restrict__ tstep,
    float* __restrict__ ctx) {
  __shared__ float pbuf[4][256];
  int lane = threadIdx.x & 31;
  int wv_ = threadIdx.x >> 5;
  int w = blockIdx.x * 4 + wv_;
  int qb = w & 63;
  int h = (w >> 6) & 15;
  int b = w >> 10;
  int m0 = qb * 16;
  int lm = lane & 15;
  int ko = (lane < 16) ? 0 : 2;
  int half8 = (lane < 16) ? 0 : 8;
  bool use_bias = (tstep[0] < 8);

  const float* Qh = q + (size_t)((b * NH + h) * SEQ) * DK;
  const float* Kh = kmat + (size_t)((b * NH + h) * SEQ) * DK;
  const float* Vh = vmat + (size_t)((b * NH + h) * SEQ) * DK;

  v2f qf[16];  // Q block 16x64 as 16 A-fragments, scale 1/sqrt(64) folded in
#pragma unroll
  for (int c = 0; c < 16; ++c)
    qf[c] = ld2(Qh + (size_t)(m0 + lm) * DK + c * 4 + ko) * 0.125f;

  float mrow[8], lrow[8];
#pragma unroll
  for (int r = 0; r < 8; ++r) { mrow[r] = NEG_INF; lrow[r] = 0.f; }
  v8f o[4] = {};

  float* pb = pbuf[wv_];

  for (int kb = 0; kb < 64; ++kb) {
    v8f s = {};
    const float* Kb = Kh + (size_t)(kb * 16 + lm) * DK + ko;
#pragma unroll
    for (int c = 0; c < 16; ++c)
      s = wmma4(qf[c], ld2(Kb + c * 4), s);

    if (use_bias && m0 < 64 && kb < 4) {
#pragma unroll
      for (int r = 0; r < 8; ++r) {
        float qg = (float)(m0 + r + half8);
        float kg = (float)(kb * 16 + lm);
        s[r] += 0.1f / (fabsf(qg - kg) + 1.0f);
      }
    }

    float p[8];
#pragma unroll
    for (int r = 0; r < 8; ++r) {
      float bm = half16_max(s[r]);
      float nm = fmaxf(mrow[r], bm);
      float corr = __expf(mrow[r] - nm);
      p[r] = __expf(s[r] - nm);
      float bs = half16_sum(p[r]);
      lrow[r] = lrow[r] * corr + bs;
      mrow[r] = nm;
#pragma unroll
      for (int t = 0; t < 4; ++t) o[t][r] *= corr;
    }

    // transpose P: C-layout (m = r+half8, n = lm) -> LDS -> A-layout fragments
#pragma unroll
    for (int r = 0; r < 8; ++r) pb[(r + half8) * 16 + lm] = p[r];
    __syncthreads();
#pragma unroll
    for (int c = 0; c < 4; ++c) {
      v2f pa;
      pa[0] = pb[lm * 16 + c * 4 + ko];
      pa[1] = pb[lm * 16 + c * 4 + ko + 1];
      const float* Vb = Vh + (size_t)(kb * 16 + c * 4 + ko) * DK + lm;
#pragma unroll
      for (int t = 0; t < 4; ++t) {
        v2f vf;
        vf[0] = Vb[t * 16];        // B[k=ko  ][n]
        vf[1] = Vb[t * 16 + DK];   // B[k=ko+1][n] (next V row)
        o[t] = wmma4(pa, vf, o[t]);
      }
    }
    __syncthreads();
  }

#pragma unroll
  for (int r = 0; r < 8; ++r) {
    float inv = 1.0f / lrow[r];
    int m = m0 + r + half8;
    float* crow = ctx + (size_t)(b * SEQ + m) * D_MODEL + h * DK;
#pragma unroll
    for (int t = 0; t < 4; ++t)
      crow[t * 16 + lm] = o[t][r] * inv;
  }
}

// ---------------- Kernel 3: attn_weights = mean_h softmax(QK^T * scale) ----------------
// One wave per (b, h, 16-query block): pass1 exact row max/sum, pass2 atomicAdd p/16.
__global__ __launch_bounds__(128) void attw_kernel(
    const float* __restrict__ q, const float* __restrict__ kmat,
    float* __restrict__ out2) {
  int lane = threadIdx.x & 31;
  int w = blockIdx.x * 4 + (threadIdx.x >> 5);
  int qb = w & 63;
  int h = (w >> 6) & 15;
  int b = w >> 10;
  int m0 = qb * 16;
  int lm = lane & 15;
  int ko = (lane < 16) ? 0 : 2;
  int half8 = (lane < 16) ? 0 : 8;
  const float* Qh = q + (size_t)((b * NH + h) * SEQ) * DK;
  const float* Kh = kmat + (size_t)((b * NH + h) * SEQ) * DK;

  v2f qf[16];
#pragma unroll
  for (int c = 0; c < 16; ++c)
    qf[c] = ld2(Qh + (size_t)(m0 + lm) * DK + c * 4 + ko) * 0.125f;

  float mrow[8], lrow[8];
#pragma unroll
  for (int r = 0; r < 8; ++r) { mrow[r] = NEG_INF; lrow[r] = 0.f; }

  for (int kb = 0; kb < 64; ++kb) {
    v8f s = {};
    const float* Kb = Kh + (size_t)(kb * 16 + lm) * DK + ko;
#pragma unroll
    for (int c = 0; c < 16; ++c)
      s = wmma4(qf[c], ld2(Kb + c * 4), s);
#pragma unroll
    for (int r = 0; r < 8; ++r) {
      float bm = half16_max(s[r]);
      float nm = fmaxf(mrow[r], bm);
      lrow[r] = lrow[r] * __expf(mrow[r] - nm) + half16_sum(__expf(s[r] - nm));
      mrow[r] = nm;
    }
  }

  float rinv[8];
#pragma unroll
  for (int r = 0; r < 8; ++r) rinv[r] = 0.0625f / lrow[r];  // /16 head mean folded in

  for (int kb = 0; kb < 64; ++kb) {
    v8f s = {};
    const float* Kb = Kh + (size_t)(kb * 16 + lm) * DK + ko;
#pragma unroll
    for (int c = 0; c < 16; ++c)
      s = wmma4(qf[c], ld2(Kb + c * 4), s);
#pragma unroll
    for (int r = 0; r < 8; ++r) {
      float pc = __expf(s[r] - mrow[r]) * rinv[r];
      int qg = m0 + r + half8;
      int kg = kb * 16 + lm;
      atomicAdd(out2 + (size_t)(b * SEQ + qg) * SEQ + kg, pc);
    }
  }
}

// ---------------- Kernel 4: output projection + residual ----------------
__global__ __launch_bounds__(256) void out_proj_kernel(
    const float* __restrict__ ctx, const float* __restrict__ Wo,
    const float* __restrict__ bo, const float* __restrict__ x,
    float* __restrict__ res) {
  int lane = threadIdx.x & 31;
  int tile = blockIdx.x * 8 + (threadIdx.x >> 5);
  int tM = tile >> 5, tN = tile & 31;
  int m0 = tM * 32, n0 = tN * 32;
  v8f acc[2][2] = {};
  gemm32x32_nt(ctx, Wo, m0, n0, lane, D_MODEL, acc);
  int lm = lane & 15;
  int half8 = (lane < 16) ? 0 : 8;
#pragma unroll
  for (int i = 0; i < 2; ++i)
#pragma unroll
    for (int j = 0; j < 2; ++j) {
      int n = n0 + j * 16 + lm;
      float bval = bo[n];
#pragma unroll
      for (int r = 0; r < 8; ++r) {
        int m = m0 + i * 16 + r + half8;
        size_t idx = (size_t)m * D_MODEL + n;
        res[idx] = acc[i][j][r] + bval + x[idx];
      }
    }
}

// ---------------- Kernel 5: row LayerNorm ----------------
__global__ __launch_bounds__(256) void ln_kernel(
    const float* __restrict__ res, const float* __restrict__ gamma,
    const float* __restrict__ beta, float* __restrict__ out) {
  __shared__ float s1[256];
  __shared__ float s2[256];
  int row = blockIdx.x, t = threadIdx.x;
  const float* r = res + (size_t)row * D_MODEL;
  float a = 0.f, bsum = 0.f;
  for (int i = t; i < D_MODEL; i += 256) {
    float vv = r[i];
    a += vv;
    bsum += vv * vv;
  }
  s1[t] = a; s2[t] = bsum;
  __syncthreads();
  for (int off = 128; off > 0; off >>= 1) {
    if (t < off) { s1[t] += s1[t + off]; s2[t] += s2[t + off]; }
    __syncthreads();
  }
  float mu = s1[0] * (1.0f / D_MODEL);
  float var = s2[0] * (1.0f / D_MODEL) - mu * mu;
  float inv = rsqrtf(var + 1e-5f);
  for (int i = t; i < D_MODEL; i += 256)
    out[(size_t)row * D_MODEL + i] = (r[i] - mu) * inv * gamma[i] + beta[i];
}

extern "C" void kernel_launch(void* const* d_in, const int* in_sizes, int n_in,
                              void* d_out, int out_size, void* d_ws, size_t ws_size,
                              hipStream_t stream) {
  (void)in_sizes; (void)n_in; (void)out_size; (void)ws_size;
  const float* x     = (const float*)d_in[0];
  const int*   tstep = (const int*)d_in[1];
  const float* Wq    = (const float*)d_in[2];
  const float* bq    = (const float*)d_in[3];
  const float* Wk    = (const float*)d_in[4];
  const float* bk    = (const float*)d_in[5];
  const float* Wv    = (const float*)d_in[6];
  const float* bv    = (const float*)d_in[7];
  const float* Wo    = (const float*)d_in[8];
  const float* bo    = (const float*)d_in[9];
  const float* gamma = (const float*)d_in[10];
  const float* beta  = (const float*)d_in[11];

  const size_t QKV = (size_t)NB * NH * SEQ * DK;  // 4,194,304 floats
  float* ws  = (float*)d_ws;
  float* q   = ws;
  float* kx  = ws + QKV;
  float* vx  = ws + 2 * QKV;
  float* ctx = ws + 3 * QKV;
  float* res = ws + 4 * QKV;

  float* out    = (float*)d_out;
  float* normed = out;                                 // [B,H,W,d] = 4M floats
  float* attw   = out + (size_t)NB * SEQ * D_MODEL;    // [B,S,S]   = 4M floats

  hipMemsetAsync(attw, 0, (size_t)NB * SEQ * SEQ * sizeof(float), stream);
  qkv_proj_kernel<<<dim3(512, 3), 256, 0, stream>>>(x, Wq, bq, Wk, bk, Wv, bv, q, kx, vx);
  attn_kernel<<<1024, 128, 0, stream>>>(q, kx, vx, tstep, ctx);
  attw_kernel<<<1024, 128, 0, stream>>>(q, kx, attw);
  out_proj_kernel<<<512, 256, 0, stream>>>(ctx, Wo, bo, x, res);
  ln_kernel<<<4096, 256, 0, stream>>>(res, gamma, beta, normed);
}